// LiquidNeuralNet_16569983828592
// MI455X (gfx1250) — compile-verified
//
#include <hip/hip_runtime.h>
#include <math.h>

// Problem constants (match reference)
#define Nn   32
#define Dm   128
#define TWO_D 256
#define Bb   64
#define Tt   128
#define DTs  0.1f
// LDS row stride (pad 128 -> 132 to avoid bank conflicts on stride-row reads)
#define SLD  132
#define BT   16   // batch tile per block

typedef float v2f __attribute__((ext_vector_type(2)));
typedef float v8f __attribute__((ext_vector_type(8)));

// ---------------------------------------------------------------------------
// Setup: prune adjacency, compute inv weight + has_parent per node
// ---------------------------------------------------------------------------
__global__ __launch_bounds__(1024) void prep_kernel(const float* __restrict__ adj,
                                                    float* __restrict__ wadj,
                                                    float* __restrict__ invw,
                                                    int* __restrict__ haspar) {
    __shared__ float sw[Nn * Nn];
    int tid = threadIdx.x;            // 1024 threads
    int i = tid >> 5, j = tid & 31;
    float a = adj[i * Nn + j];
    float w = (a > 0.01f && i != j) ? a : 0.0f;
    sw[i * Nn + j] = w;
    wadj[i * Nn + j] = w;
    __syncthreads();
    if (j == 0) {
        float tw = 0.0f;
        for (int jj = 0; jj < Nn; ++jj) tw += sw[i * Nn + jj];
        haspar[i] = (tw > 0.0f) ? 1 : 0;
        invw[i] = 1.0f / fmaxf(tw, 1e-12f);
    }
}

__global__ __launch_bounds__(256) void zero_kernel(float* __restrict__ p, int n) {
    int i = blockIdx.x * 256 + threadIdx.x;
    if (i < n) p[i] = 0.0f;
}

// ---------------------------------------------------------------------------
// One RK4 substep for node n, batch rows [b0, b0+16), with the parent
// reduction fused in (reads previous-substep state of all nodes; the kernel
// boundary between substeps provides the cross-workgroup ordering).
// grid = (B/16, N), block = 256 threads (8 waves)
// Each stage: 24 16x16 output tiles (3 matrices x 8 col tiles) via
// V_WMMA_F32_16X16X4_F32, K = 256 over [parent | cur_state].
// ---------------------------------------------------------------------------
__global__ __launch_bounds__(256) void rk4_kernel(float* __restrict__ state,
                                                  const float* __restrict__ emb,
                                                  const float* __restrict__ wadj,
                                                  const float* __restrict__ invw,
                                                  const int* __restrict__ haspar,
                                                  const float* __restrict__ W_tau,
                                                  const float* __restrict__ b_tau,
                                                  const float* __restrict__ W_f,
                                                  const float* __restrict__ b_f,
                                                  const float* __restrict__ W_gate,
                                                  const float* __restrict__ b_gate,
                                                  const float* __restrict__ bias,
                                                  float* __restrict__ out,
                                                  int t, int write_out) {
    __shared__ float sParent[BT * SLD];
    __shared__ float sState [BT * SLD];
    __shared__ float sCur   [BT * SLD];
    __shared__ float sKsum  [BT * SLD];
    __shared__ float sAct[3][BT * SLD];   // tau, f, gate
    __shared__ float sWrow[Nn];

    const int b0 = blockIdx.x * BT;
    const int n  = blockIdx.y;
    const int tid  = threadIdx.x;
    const int wave = tid >> 5;
    const int lane = tid & 31;
    const int c  = lane & 15;             // A row / B col within tile
    const int hi = lane >> 4;             // K sub-offset selector

    const int   hasp = haspar[n];         // uniform
    const float inv  = invw[n];           // uniform

    if (tid < Nn) sWrow[tid] = wadj[n * Nn + tid];
    __syncthreads();

    // Fused parent + state load into LDS
    for (int i = tid; i < BT * Dm; i += 256) {
        int r = i >> 7, d = i & (Dm - 1);
        int b = b0 + r;
        const float* erow = emb + ((size_t)b * Tt + t) * (Nn * Dm) + d;  // [j] -> erow[j*Dm]
        const float* srow = state + (size_t)b * (Nn * Dm) + d;           // [j] -> srow[j*Dm]
        float st = srow[n * Dm];                                          // own node state
        float p;
        if (hasp) {
            float acc = 0.0f;
            #pragma unroll 8
            for (int j = 0; j < Nn; ++j)
                acc += sWrow[j] * (erow[j * Dm] + srow[j * Dm]);
            p = acc * (0.5f * inv);       // mixed = 0.5*(emb+state), factored out
        } else {
            p = 0.1f * erow[n * Dm];
        }
        sState [r * SLD + d] = st;
        sCur   [r * SLD + d] = st;
        sParent[r * SLD + d] = p;
        sKsum  [r * SLD + d] = 0.0f;
    }

    const float incc[4] = {0.5f * DTs, 0.5f * DTs, DTs, 0.0f};
    const float wsum[4] = {1.0f, 2.0f, 2.0f, 1.0f};

    for (int s = 0; s < 4; ++s) {
        __syncthreads();   // sCur / sParent ready

        // ---- GEMM phase: each wave computes 3 tiles (one per matrix) ----
        for (int tt = wave; tt < 24; tt += 8) {
            const int mat = tt >> 3;      // 0=tau, 1=f, 2=gate
            const int col = tt & 7;
            const int d = col * 16 + c;
            const float* W  = (mat == 0 ? W_tau : (mat == 1 ? W_f : W_gate))
                              + (size_t)n * TWO_D * Dm + d;     // W[k][d] = W[k*Dm]
            const float* bv = (mat == 0 ? b_tau : (mat == 1 ? b_f : b_gate)) + n * Dm;

            v8f C = {};
            // K = 0..127 : A from parent
            #pragma unroll 8
            for (int k0 = 0; k0 < 128; k0 += 4) {
                const int kb = k0 + 2 * hi;
                v2f a, b;
                a.x = sParent[c * SLD + kb];
                a.y = sParent[c * SLD + kb + 1];
                b.x = W[(size_t)kb * Dm];
                b.y = W[(size_t)(kb + 1) * Dm];
                C = __builtin_amdgcn_wmma_f32_16x16x4_f32(
                        false, a, false, b, (short)0, C, false, false);
            }
            // K = 128..255 : A from current stage state
            #pragma unroll 8
            for (int k0 = 0; k0 < 128; k0 += 4) {
                const int kb = k0 + 2 * hi;
                v2f a, b;
                a.x = sCur[c * SLD + kb];
                a.y = sCur[c * SLD + kb + 1];
                b.x = W[(size_t)(128 + kb) * Dm];
                b.y = W[(size_t)(129 + kb) * Dm];
                C = __builtin_amdgcn_wmma_f32_16x16x4_f32(
                        false, a, false, b, (short)0, C, false, false);
            }

            const float bb = bv[d];
            const float extra = (mat == 1) ? bias[n * Dm + d] : 0.0f;
            float* dst = &sAct[mat][0];
            #pragma unroll
            for (int v = 0; v < 8; ++v) {
                float x = C[v] + bb;
                float y;
                if (mat == 0)       y = 1.0f / (1.0f + __expf(-x)) + 0.1f;  // tau
                else if (mat == 1)  y = tanhf(x) + extra;                    // f
                else                y = 1.0f / (1.0f + __expf(-x));          // gate
                dst[(v + 8 * hi) * SLD + d] = y;   // row = M, col = d
            }
        }
        __syncthreads();   // activations written + last sCur reads done

        // ---- elementwise RK4 combine ----
        for (int i = tid; i < BT * Dm; i += 256) {
            int r = i >> 7, d = i & (Dm - 1);
            int si = r * SLD + d;
            float tau  = sAct[0][si];
            float fv   = sAct[1][si];
            float gate = sAct[2][si];
            float k = gate * (fv - sCur[si]) / tau;
            sKsum[si] += wsum[s] * k;
            if (s < 3) sCur[si] = sState[si] + incc[s] * k;
        }
    }
    __syncthreads();

    // ---- write back new state (+ trajectory on last substep of timestep) ----
    for (int i = tid; i < BT * Dm; i += 256) {
        int r = i >> 7, d = i & (Dm - 1);
        float ns = sState[r * SLD + d] + (DTs / 6.0f) * sKsum[r * SLD + d];
        ns = fminf(5.0f, fmaxf(-5.0f, ns));
        int g = ((b0 + r) * Nn + n) * Dm + d;
        state[g] = ns;
        if (write_out)
            out[(((size_t)(b0 + r) * Tt + t) * Nn + n) * Dm + d] = ns;
    }
}

// ---------------------------------------------------------------------------
extern "C" void kernel_launch(void* const* d_in, const int* in_sizes, int n_in,
                              void* d_out, int out_size, void* d_ws, size_t ws_size,
                              hipStream_t stream) {
    const float* emb    = (const float*)d_in[0];
    const float* adj    = (const float*)d_in[1];
    const float* W_tau  = (const float*)d_in[2];
    const float* b_tau  = (const float*)d_in[3];
    const float* W_f    = (const float*)d_in[4];
    const float* b_f    = (const float*)d_in[5];
    const float* W_gate = (const float*)d_in[6];
    const float* b_gate = (const float*)d_in[7];
    const float* bias   = (const float*)d_in[8];
    float* out = (float*)d_out;

    const int STATE_ELEMS = Bb * Nn * Dm;          // 262144
    float* state  = (float*)d_ws;
    float* wadj   = state + STATE_ELEMS;           // 32*32
    float* invw   = wadj + Nn * Nn;                // 32
    int*   haspar = (int*)(invw + Nn);             // 32

    zero_kernel<<<(STATE_ELEMS + 255) / 256, 256, 0, stream>>>(state, STATE_ELEMS);
    prep_kernel<<<1, 1024, 0, stream>>>(adj, wadj, invw, haspar);

    for (int t = 0; t < Tt; ++t) {
        for (int s = 0; s < 4; ++s) {
            rk4_kernel<<<dim3(Bb / BT, Nn), 256, 0, stream>>>(
                state, emb, wadj, invw, haspar,
                W_tau, b_tau, W_f, b_f, W_gate, b_gate, bias,
                out, t, (s == 3) ? 1 : 0);
        }
    }
}